// GS_GraphSAT_54829552500983
// MI455X (gfx1250) — compile-verified
//
#include <hip/hip_runtime.h>
#include <math.h>

typedef float v2f __attribute__((ext_vector_type(2)));
typedef float v8f __attribute__((ext_vector_type(8)));

#define NEGBIG (-1.0e30f)

static __device__ inline v8f wmma4(v2f a, v2f b, v8f c) {
  // D = A(16x4,f32) * B(4x16,f32) + C(16x16,f32)
  return __builtin_amdgcn_wmma_f32_16x16x4_f32(false, a, false, b, (short)0, c,
                                               false, false);
}

// ---------------------------------------------------------------------------
// Generic single-wave WMMA f32 GEMM tile: C[z] = A[z] @ B[z] (+ bias[z])
// grid = (Nc/16, M/16, Z), block = 32
// ---------------------------------------------------------------------------
__global__ __launch_bounds__(32) void gemm_wave(
    const float* __restrict__ A, const float* __restrict__ B,
    const float* __restrict__ bias, float* __restrict__ C,
    int M, int K, int Nc, long As, long Bs, long bs, long Cs)
{
  const int z = blockIdx.z;
  A += (long)z * As;
  B += (long)z * Bs;
  C += (long)z * Cs;
  const int row0 = blockIdx.y * 16, col0 = blockIdx.x * 16;
  const int lane = threadIdx.x & 31, half = lane >> 4, l15 = lane & 15;
  v8f acc = {};
  for (int k = 0; k < K; k += 4) {
    v2f a, b;
    a.x = A[(long)(row0 + l15) * K + k + 2 * half];
    a.y = A[(long)(row0 + l15) * K + k + 2 * half + 1];
    b.x = B[(long)(k + 2 * half) * Nc + col0 + l15];
    b.y = B[(long)(k + 2 * half + 1) * Nc + col0 + l15];
    acc = wmma4(a, b, acc);
  }
  float bv = 0.f;
  if (bias) bv = (bias + (long)z * bs)[col0 + l15];
#pragma unroll
  for (int r = 0; r < 8; ++r)
    C[(long)(row0 + r + 8 * half) * Nc + col0 + l15] = acc[r] + bv;
}

// ---------------------------------------------------------------------------
// f1[i] = Wh[i,:] . a[0:F], f2[i] = Wh[i,:] . a[F:2F]   (per z)
// ---------------------------------------------------------------------------
__global__ void fvec(const float* __restrict__ Wh, const float* __restrict__ a,
                     float* __restrict__ f1, float* __restrict__ f2,
                     int N, int F, long WhS, long aS, long fS)
{
  const int z = blockIdx.y;
  Wh += (long)z * WhS;
  a += (long)z * aS;
  f1 += (long)z * fS;
  f2 += (long)z * fS;
  int i = blockIdx.x * blockDim.x + threadIdx.x;
  if (i >= N) return;
  float s1 = 0.f, s2 = 0.f;
  for (int c = 0; c < F; ++c) {
    float w = Wh[(long)i * F + c];
    s1 += w * a[c];
    s2 += w * a[F + c];
  }
  f1[i] = s1;
  f2[i] = s2;
}

// ---------------------------------------------------------------------------
// Fused flash-style GraphSelfAttention layer body.
//   F: out_features (32 hidden, 64 output); inner heads = 4, d = F/4.
// One wave handles 16 rows; two passes over all j tiles; no NxN materialized.
// grid = (1, N/16, heads), block = 32
// ---------------------------------------------------------------------------
template <int F>
__global__ __launch_bounds__(32) void attn_kernel(
    const float* __restrict__ Wh, const float* __restrict__ qkv,
    const float* __restrict__ f1, const float* __restrict__ f2,
    const int* __restrict__ adj, float* __restrict__ out,
    int N, long WhS, long qkvS, long fS,
    int outStride, int colOffPerHead, int eluCount)
{
  constexpr int d = F / 4;    // per inner-head dim (8 or 16)
  constexpr int NT = F / 16;  // feature column tiles (2 or 4)
  constexpr int KS = d / 4;   // wmma K-steps for QK^T

  const int h = blockIdx.z;
  const int i0 = blockIdx.y * 16;
  Wh += (long)h * WhS;
  const float* q = qkv + (long)h * qkvS;
  const float* kb = q + F;
  const float* vb = q + 2 * F;
  f1 += (long)h * fS;
  f2 += (long)h * fS;

  const int lane = threadIdx.x & 31, half = lane >> 4, l15 = lane & 15;
  const int ldq = 3 * F;
  const float scale = rsqrtf((float)F);

  __shared__ float Pl[4][16][16];  // inner-head probability tiles
  __shared__ float Wl[16][16];     // outer attention weight tile
  __shared__ float S0[16][F];      // Wh0 staging
  __shared__ float S1[16][F];      // outer accumulator staging
  __shared__ float SZ[16];         // outer normalizers

  // Preload Q as A-fragments (row = l15, K-chunk selected by half).
  float qa0[4][KS], qa1[4][KS];
#pragma unroll
  for (int hh = 0; hh < 4; ++hh)
#pragma unroll
    for (int ks = 0; ks < KS; ++ks) {
      int kc = hh * d + 4 * ks + 2 * half;
      qa0[hh][ks] = q[(long)(i0 + l15) * ldq + kc];
      qa1[hh][ks] = q[(long)(i0 + l15) * ldq + kc + 1];
    }
  float f1v[8];
#pragma unroll
  for (int r = 0; r < 8; ++r) f1v[r] = f1[i0 + r + 8 * half];

  // ---------------- Pass A: inner softmax row stats (online) --------------
  float im[4][8], is[4][8];
#pragma unroll
  for (int hh = 0; hh < 4; ++hh)
#pragma unroll
    for (int r = 0; r < 8; ++r) { im[hh][r] = NEGBIG; is[hh][r] = 0.f; }

  const int NJT = N / 16;
  for (int jt = 0; jt < NJT; ++jt) {
    const int j0 = jt * 16;
#pragma unroll
    for (int hh = 0; hh < 4; ++hh) {
      v8f acc = {};
#pragma unroll
      for (int ks = 0; ks < KS; ++ks) {
        int kc = hh * d + 4 * ks + 2 * half;
        v2f a, b;
        a.x = qa0[hh][ks]; a.y = qa1[hh][ks];
        b.x = kb[(long)(j0 + l15) * ldq + kc];
        b.y = kb[(long)(j0 + l15) * ldq + kc + 1];
        acc = wmma4(a, b, acc);
      }
#pragma unroll
      for (int r = 0; r < 8; ++r) {
        float sv = acc[r] * scale;
        float mn = fmaxf(im[hh][r], sv);
        is[hh][r] = is[hh][r] * __expf(im[hh][r] - mn) + __expf(sv - mn);
        im[hh][r] = mn;
      }
    }
  }
  // Merge stats across the 16 column lanes (xor stays within each half).
#pragma unroll
  for (int mk = 1; mk <= 8; mk <<= 1)
#pragma unroll
    for (int hh = 0; hh < 4; ++hh)
#pragma unroll
      for (int r = 0; r < 8; ++r) {
        float om = __shfl_xor(im[hh][r], mk, 32);
        float os = __shfl_xor(is[hh][r], mk, 32);
        float mn = fmaxf(im[hh][r], om);
        is[hh][r] = is[hh][r] * __expf(im[hh][r] - mn) + os * __expf(om - mn);
        im[hh][r] = mn;
      }

  // ---------------- Pass B: probabilities + accumulations ------------------
  v8f w0acc[4];
  v8f oacc[NT];
#pragma unroll
  for (int hh = 0; hh < 4; ++hh) w0acc[hh] = (v8f){};
#pragma unroll
  for (int t = 0; t < NT; ++t) oacc[t] = (v8f){};
  float mo[8], zo[8];
#pragma unroll
  for (int r = 0; r < 8; ++r) { mo[r] = NEGBIG; zo[r] = 0.f; }

  for (int jt = 0; jt < NJT; ++jt) {
    const int j0 = jt * 16;
    float dsum[8];
#pragma unroll
    for (int r = 0; r < 8; ++r) dsum[r] = 0.f;

    // inner dots tiles
#pragma unroll
    for (int hh = 0; hh < 4; ++hh) {
      v8f acc = {};
#pragma unroll
      for (int ks = 0; ks < KS; ++ks) {
        int kc = hh * d + 4 * ks + 2 * half;
        v2f a, b;
        a.x = qa0[hh][ks]; a.y = qa1[hh][ks];
        b.x = kb[(long)(j0 + l15) * ldq + kc];
        b.y = kb[(long)(j0 + l15) * ldq + kc + 1];
        acc = wmma4(a, b, acc);
      }
#pragma unroll
      for (int r = 0; r < 8; ++r) {
        float p = __expf(acc[r] * scale - im[hh][r]) / is[hh][r];
        dsum[r] += p;
        Pl[hh][r + 8 * half][l15] = p;
      }
    }

    // outer logits + online softmax bookkeeping
    const float f2v = f2[j0 + l15];
    float lw[8], msb[8];
#pragma unroll
    for (int r = 0; r < 8; ++r) {
      int row = i0 + r + 8 * half;
      float e = f1v[r] + f2v;
      e = (e > 0.f) ? e : 0.2f * e;  // leaky relu
      float lg = e + dsum[r];
      bool msk = adj[(long)row * N + j0 + l15] > 0;
      msb[r] = msk ? 1.f : 0.f;
      lw[r] = msk ? lg : NEGBIG;
    }
    float tmax[8];
#pragma unroll
    for (int r = 0; r < 8; ++r) tmax[r] = lw[r];
#pragma unroll
    for (int mk = 1; mk <= 8; mk <<= 1)
#pragma unroll
      for (int r = 0; r < 8; ++r)
        tmax[r] = fmaxf(tmax[r], __shfl_xor(tmax[r], mk, 32));
#pragma unroll
    for (int r = 0; r < 8; ++r) {
      float mn = fmaxf(mo[r], tmax[r]);
      float fr = __expf(mo[r] - mn);  // (NEGBIG-NEGBIG)->1, harmless: zo/acc==0
      zo[r] *= fr;
#pragma unroll
      for (int t = 0; t < NT; ++t) oacc[t][r] *= fr;
      mo[r] = mn;
      float wt = msb[r] * __expf(lw[r] - mn);
      zo[r] += wt;
      Wl[r + 8 * half][l15] = wt;
    }
    __syncthreads();

    // Wh0 += dots @ V  (per inner head; cols >= d padded with zero)
#pragma unroll
    for (int hh = 0; hh < 4; ++hh) {
      v8f acc = w0acc[hh];
#pragma unroll
      for (int ks = 0; ks < 4; ++ks) {
        int kc = 4 * ks + 2 * half;
        v2f a, b;
        a.x = Pl[hh][l15][kc];
        a.y = Pl[hh][l15][kc + 1];
        float b0 = 0.f, b1 = 0.f;
        if (l15 < d) {
          b0 = vb[(long)(j0 + kc) * ldq + hh * d + l15];
          b1 = vb[(long)(j0 + kc + 1) * ldq + hh * d + l15];
        }
        b.x = b0; b.y = b1;
        acc = wmma4(a, b, acc);
      }
      w0acc[hh] = acc;
    }
    // outer: acc += w @ Wh
#pragma unroll
    for (int t = 0; t < NT; ++t) {
      v8f acc = oacc[t];
#pragma unroll
      for (int ks = 0; ks < 4; ++ks) {
        int kc = 4 * ks + 2 * half;
        v2f a, b;
        a.x = Wl[l15][kc];
        a.y = Wl[l15][kc + 1];
        b.x = Wh[(long)(j0 + kc) * F + t * 16 + l15];
        b.y = Wh[(long)(j0 + kc + 1) * F + t * 16 + l15];
        acc = wmma4(a, b, acc);
      }
      oacc[t] = acc;
    }
    __syncthreads();
  }

  // ---------------- Finalize ----------------------------------------------
#pragma unroll
  for (int mk = 1; mk <= 8; mk <<= 1)
#pragma unroll
    for (int r = 0; r < 8; ++r) zo[r] += __shfl_xor(zo[r], mk, 32);

#pragma unroll
  for (int hh = 0; hh < 4; ++hh)
#pragma unroll
    for (int r = 0; r < 8; ++r)
      if (l15 < d) S0[r + 8 * half][hh * d + l15] = w0acc[hh][r];
#pragma unroll
  for (int t = 0; t < NT; ++t)
#pragma unroll
    for (int r = 0; r < 8; ++r)
      S1[r + 8 * half][t * 16 + l15] = oacc[t][r];
  if (l15 == 0) {
#pragma unroll
    for (int r = 0; r < 8; ++r) SZ[r + 8 * half] = zo[r];
  }
  __syncthreads();

  if (lane < 16) {
    const int ri = lane;
    float mx = NEGBIG;
    for (int c = 0; c < F; ++c) mx = fmaxf(mx, S0[ri][c]);
    float sm = 0.f;
    for (int c = 0; c < F; ++c) sm += __expf(S0[ri][c] - mx);
    const float smInv = 1.f / sm;
    const float zInv = 1.f / fmaxf(SZ[ri], 1e-30f);
    for (int c = 0; c < F; ++c) {
      float val = S1[ri][c] * zInv + __expf(S0[ri][c] - mx) * smInv;
      float r1 = (val > 0.f) ? val : (__expf(val) - 1.f);
      if (eluCount == 2) r1 = (r1 > 0.f) ? r1 : (__expf(r1) - 1.f);
      out[(long)(i0 + ri) * outStride + colOffPerHead * h + c] = r1;
    }
  }
}

// ---------------------------------------------------------------------------
extern "C" void kernel_launch(void* const* d_in, const int* in_sizes, int n_in,
                              void* d_out, int out_size, void* d_ws,
                              size_t ws_size, hipStream_t stream)
{
  const float* x      = (const float*)d_in[0];
  const int*   adj    = (const int*)d_in[1];
  const float* W_h    = (const float*)d_in[2];
  const float* linW_h = (const float*)d_in[3];
  const float* linb_h = (const float*)d_in[4];
  const float* qkvW_h = (const float*)d_in[5];
  const float* qkvb_h = (const float*)d_in[6];
  const float* a_h    = (const float*)d_in[7];
  const float* W_o    = (const float*)d_in[8];
  const float* linW_o = (const float*)d_in[9];
  const float* linb_o = (const float*)d_in[10];
  const float* qkvW_o = (const float*)d_in[11];
  const float* qkvb_o = (const float*)d_in[12];
  const float* a_o    = (const float*)d_in[13];
  float* out = (float*)d_out;

  const int N = 3072, NF = 128, FH = 32, FO = 64;
  float* ws   = (float*)d_ws;
  float* T4   = ws;                     // 4*N*FH
  float* Wh4  = T4 + 4 * N * FH;        // 4*N*FH
  float* qkv4 = Wh4 + 4 * N * FH;       // 4*N*3*FH
  float* f1_4 = qkv4 + 4 * N * 3 * FH;  // 4*N
  float* f2_4 = f1_4 + 4 * N;           // 4*N
  float* x1   = f2_4 + 4 * N;           // N*128
  float* To   = x1 + N * 128;           // N*FO
  float* Who  = To + N * FO;            // N*FO
  float* qkvo = Who + N * FO;           // N*3*FO
  float* f1o  = qkvo + N * 3 * FO;      // N
  float* f2o  = f1o + N;                // N

  dim3 b32(32);

  // ---- hidden layer (4 outer heads batched over blockIdx.z) ----
  gemm_wave<<<dim3(FH / 16, N / 16, 4), b32, 0, stream>>>(
      x, W_h, nullptr, T4, N, NF, FH, 0L, (long)NF * FH, 0L, (long)N * FH);
  gemm_wave<<<dim3(FH / 16, N / 16, 4), b32, 0, stream>>>(
      T4, linW_h, linb_h, Wh4, N, FH, FH, (long)N * FH, (long)FH * FH,
      (long)FH, (long)N * FH);
  gemm_wave<<<dim3(3 * FH / 16, N / 16, 4), b32, 0, stream>>>(
      Wh4, qkvW_h, qkvb_h, qkv4, N, FH, 3 * FH, (long)N * FH,
      (long)FH * 3 * FH, (long)3 * FH, (long)N * 3 * FH);
  fvec<<<dim3((N + 127) / 128, 4), dim3(128), 0, stream>>>(
      Wh4, a_h, f1_4, f2_4, N, FH, (long)N * FH, (long)2 * FH, (long)N);
  attn_kernel<32><<<dim3(1, N / 16, 4), b32, 0, stream>>>(
      Wh4, qkv4, f1_4, f2_4, adj, x1, N, (long)N * FH, (long)N * 3 * FH,
      (long)N, 128, 32, 2);

  // ---- output layer ----
  gemm_wave<<<dim3(FO / 16, N / 16, 1), b32, 0, stream>>>(
      x1, W_o, nullptr, To, N, 128, FO, 0L, 0L, 0L, 0L);
  gemm_wave<<<dim3(FO / 16, N / 16, 1), b32, 0, stream>>>(
      To, linW_o, linb_o, Who, N, FO, FO, 0L, 0L, 0L, 0L);
  gemm_wave<<<dim3(3 * FO / 16, N / 16, 1), b32, 0, stream>>>(
      Who, qkvW_o, qkvb_o, qkvo, N, FO, 3 * FO, 0L, 0L, 0L, 0L);
  fvec<<<dim3((N + 127) / 128, 1), dim3(128), 0, stream>>>(
      Who, a_o, f1o, f2o, N, FO, 0L, 0L, 0L);
  attn_kernel<64><<<dim3(1, N / 16, 1), b32, 0, stream>>>(
      Who, qkvo, f1o, f2o, adj, out, N, 0L, 0L, 0L, 64, 0, 1);
}